// MonomialMatrixModule_80393197847148
// MI455X (gfx1250) — compile-verified
//
#include <hip/hip_runtime.h>

// ---------------------------------------------------------------------------
// out[j,k] = sum_i coeffs[j,k,i] * v[i],  J=K=2048, M=84 monomials of x (6 vars,
// order 3).  1.41 GB streamed / 0.70 GFLOP -> pure HBM-bound (~60us floor at
// 23.3 TB/s).  Reduction routed through V_WMMA_F32_16X16X4_F32: one wave per
// 16-row tile, 21 K=4 chunks, A = broadcast slice of v (built from scalar
// loads + cndmask, zero VMEM), B = per-lane 8B non-temporal global loads.
// ---------------------------------------------------------------------------

typedef __attribute__((ext_vector_type(2))) float v2f;
typedef __attribute__((ext_vector_type(8))) float v8f;

#define NVARS  6
#define MMONO  84   // 1 + 6 + 21 + 56
#define NCHUNK 21   // 84 / 4

// ---- Kernel 1: monomial vector v[84] in itertools order ------------------
__global__ void monomial_vec_kernel(const float* __restrict__ x,
                                    float* __restrict__ v) {
  if (threadIdx.x == 0) {
    float xl[NVARS];
#pragma unroll
    for (int i = 0; i < NVARS; ++i) xl[i] = x[i];
    int idx = 0;
    v[idx++] = 1.0f;                                   // constant term
#pragma unroll
    for (int a = 0; a < NVARS; ++a) v[idx++] = xl[a];  // degree 1
#pragma unroll
    for (int a = 0; a < NVARS; ++a)                    // degree 2 (CWR order)
#pragma unroll
      for (int b = a; b < NVARS; ++b) v[idx++] = xl[a] * xl[b];
#pragma unroll
    for (int a = 0; a < NVARS; ++a)                    // degree 3 (CWR order)
#pragma unroll
      for (int b = a; b < NVARS; ++b)
#pragma unroll
        for (int c = b; c < NVARS; ++c) v[idx++] = xl[a] * xl[b] * xl[c];
  }
}

// ---- Kernel 2: streaming mat-vec via fp32 WMMA ---------------------------
// Tile = 16 consecutive rows per wave.  Chunk c covers K = 4c..4c+3.
//   A (16x4, all rows identical = v slice):
//     lane half hi: VGPR0 = v[4c+2hi], VGPR1 = v[4c+2hi+1]
//   B (4x16, coeffs):
//     lane (n, hi):  VGPR0 = coeffs[row n][4c+2hi], VGPR1 = [..+1]
//   D: every row equals out[row n] at column n -> lanes 0..15 store acc[0].
__global__ __launch_bounds__(256) void monomial_matvec_wmma(
    const float* __restrict__ coeffs, const float* __restrict__ v,
    float* __restrict__ out, long long nrows) {
  const int lane = (int)(threadIdx.x & 31u);
  const int n    = lane & 15;   // row-in-tile == output column
  const int hi   = lane >> 4;   // 0: lanes 0-15, 1: lanes 16-31

  const long long tile =
      (long long)blockIdx.x * (blockDim.x >> 5) + (threadIdx.x >> 5);
  const long long rowbase = tile * 16;
  if (rowbase >= nrows) return;  // wave-uniform: EXEC stays all-ones below

  long long row = rowbase + n;
  if (row >= nrows) row = nrows - 1;  // clamp ragged-tail reads (unused here)

  // Per-lane base pointer; chunk offset is a 16-byte immediate (8B aligned:
  // row*84 and 2*hi are even element offsets).
  const float* bptr = coeffs + row * (long long)MMONO + (hi << 1);

  v8f acc = {0.f, 0.f, 0.f, 0.f, 0.f, 0.f, 0.f, 0.f};

#pragma unroll
  for (int c = 0; c < NCHUNK; ++c) {
    // B operand: one global_load_b64 per lane, non-temporal (stream, no reuse)
    v2f b = __builtin_nontemporal_load((const v2f*)(bptr + 4 * c));

    // A operand: wave-uniform scalar loads of v + per-half select
    float ve0 = v[4 * c + 0];
    float ve1 = v[4 * c + 1];
    float vo0 = v[4 * c + 2];
    float vo1 = v[4 * c + 3];
    v2f a;
    a.x = hi ? vo0 : ve0;
    a.y = hi ? vo1 : ve1;

    // D = A x B + acc   (8 args: neg_a, A, neg_b, B, c_mod, C, reuse_a, reuse_b)
    acc = __builtin_amdgcn_wmma_f32_16x16x4_f32(
        false, a, false, b, (short)0, acc, false, false);
  }

  if (hi == 0 && (rowbase + n) < nrows) {
    out[rowbase + n] = acc[0];  // coalesced 16-float store per tile
  }
}

// ---- Launch ---------------------------------------------------------------
extern "C" void kernel_launch(void* const* d_in, const int* in_sizes, int n_in,
                              void* d_out, int out_size, void* d_ws,
                              size_t ws_size, hipStream_t stream) {
  const float* x      = (const float*)d_in[0];  // (6,)  float32
  const float* coeffs = (const float*)d_in[1];  // (J,K,84) float32
  float* out = (float*)d_out;                   // (J,K) float32
  float* v   = (float*)d_ws;                    // 84 floats of scratch

  monomial_vec_kernel<<<1, 32, 0, stream>>>(x, v);

  const long long nrows = (long long)out_size;          // J*K = 4,194,304
  const long long tiles = (nrows + 15) / 16;            // 16 rows per wave
  const int waves_per_block = 8;                        // 256 threads, wave32
  const long long blocks =
      (tiles + waves_per_block - 1) / waves_per_block;  // 32,768

  monomial_matvec_wmma<<<(int)blocks, waves_per_block * 32, 0, stream>>>(
      coeffs, v, out, nrows);
}